// SqueezeAndExcitation_4879082848777
// MI455X (gfx1250) — compile-verified
//
#include <hip/hip_runtime.h>
#include <hip/hip_bf16.h>

// Squeeze-and-Excitation, MI455X (gfx1250, wave32).
//   x: [64, 384, 56, 56] f32; planes = 64*384 = 24576; HW = 3136 (784 float4)
//   Pure HBM-bound: ~924 MB traffic -> ~40us @ 23.3 TB/s. MLP done with native
//   fp32 WMMA (v_wmma_f32_16x16x4_f32) to keep full precision at zero cost.

typedef __attribute__((ext_vector_type(2))) float v2f;
typedef __attribute__((ext_vector_type(4))) float v4f;
typedef __attribute__((ext_vector_type(8))) float v8f;

#define NB    64
#define NC    384
#define NSQ   96
#define HW    3136
#define HW4   784          // HW / 4
#define NPLANES (NB * NC)  // 24576
#define SSTRIDE 97         // LDS row stride for s-matrix (gcd(97,64)=1 -> no bank conflicts)

// ---------------------------------------------------------------------------
// Kernel 1: global average pool. One 256-thread block per (b,c) plane.
// ---------------------------------------------------------------------------
__global__ __launch_bounds__(256) void se_pool(const float* __restrict__ x,
                                               float* __restrict__ y) {
    const int plane = blockIdx.x;
    const v4f* xp = (const v4f*)(x + (size_t)plane * HW);
    const int t = threadIdx.x;

    float s = 0.0f;
    v4f a;
    a = xp[t];        s += a.x + a.y + a.z + a.w;
    a = xp[t + 256];  s += a.x + a.y + a.z + a.w;
    a = xp[t + 512];  s += a.x + a.y + a.z + a.w;
    if (t < HW4 - 768) {  // 784 - 768 = 16 remaining vectors
        a = xp[t + 768];
        s += a.x + a.y + a.z + a.w;
    }

    // wave32 butterfly reduction
    #pragma unroll
    for (int off = 16; off > 0; off >>= 1)
        s += __shfl_xor(s, off, 32);

    __shared__ float wsum[8];
    if ((t & 31) == 0) wsum[t >> 5] = s;
    __syncthreads();
    if (t == 0) {
        float tot = 0.0f;
        #pragma unroll
        for (int i = 0; i < 8; ++i) tot += wsum[i];
        y[plane] = tot * (1.0f / (float)HW);
    }
}

// ---------------------------------------------------------------------------
// Kernel 2: tiny MLP with fp32 WMMA. Single block, 32 waves.
//   Stage 1 (reduce):  s[64,96]  = swish(y[64,384] @ w1[96,384]^T + b1)
//   Stage 2 (excite):  g[64,384] = sigmoid(s[64,96] @ w2[384,96]^T + b2)
// A-frag lane map: M = lane%16, K = vgpr + 2*(lane/16)
// B-frag lane map: N = lane%16, K = vgpr + 2*(lane/16)   (B[k][n] = W[n][k])
// D lane map:      M = vgpr + 8*(lane/16), N = lane%16
// ---------------------------------------------------------------------------
__device__ __forceinline__ float sigmoidf_(float v) {
    return 1.0f / (1.0f + __expf(-v));
}

__global__ __launch_bounds__(1024) void se_mlp(const float* __restrict__ y,
                                               const float* __restrict__ w1,
                                               const float* __restrict__ b1,
                                               const float* __restrict__ w2,
                                               const float* __restrict__ b2,
                                               float* __restrict__ g) {
    __shared__ float s_lds[NB * SSTRIDE];

    const int tid  = threadIdx.x;
    const int wave = tid >> 5;
    const int lane = tid & 31;
    const int half = lane >> 4;   // 0 or 1
    const int l16  = lane & 15;
    const int kb   = half * 2;    // K base offset for this half-wave

    // ---- Stage 1: 4x6 = 24 output tiles, one wave each -------------------
    if (wave < 24) {
        const int m0 = (wave / 6) * 16;   // batch tile
        const int n0 = (wave % 6) * 16;   // squeeze-channel tile
        v8f acc = {};
        const float* Ya = y  + (size_t)(m0 + l16) * NC;   // A rows (batch)
        const float* Wa = w1 + (size_t)(n0 + l16) * NC;   // B cols (w1 row-major [s,c])
        for (int k0 = 0; k0 < NC; k0 += 4) {
            v2f a, b;
            a.x = Ya[k0 + kb];  a.y = Ya[k0 + kb + 1];
            b.x = Wa[k0 + kb];  b.y = Wa[k0 + kb + 1];
            acc = __builtin_amdgcn_wmma_f32_16x16x4_f32(
                false, a, false, b, (short)0, acc, false, false);
        }
        const float bias = b1[n0 + l16];
        #pragma unroll
        for (int v = 0; v < 8; ++v) {
            const int m = m0 + v + 8 * half;
            float t = acc[v] + bias;
            s_lds[m * SSTRIDE + (n0 + l16)] = t * sigmoidf_(t);  // Swish
        }
    }
    __syncthreads();

    // ---- Stage 2: 4x24 = 96 output tiles, 3 per wave ----------------------
    #pragma unroll
    for (int t3 = 0; t3 < 3; ++t3) {
        const int tile = wave * 3 + t3;      // 0..95
        const int m0 = (tile / 24) * 16;     // batch tile
        const int n0 = (tile % 24) * 16;     // out-channel tile
        v8f acc = {};
        const float* Sa = s_lds + (size_t)(m0 + l16) * SSTRIDE;
        const float* Wb = w2 + (size_t)(n0 + l16) * NSQ;   // w2 row-major [c,s]
        for (int k0 = 0; k0 < NSQ; k0 += 4) {
            v2f a, b;
            a.x = Sa[k0 + kb];  a.y = Sa[k0 + kb + 1];
            b.x = Wb[k0 + kb];  b.y = Wb[k0 + kb + 1];
            acc = __builtin_amdgcn_wmma_f32_16x16x4_f32(
                false, a, false, b, (short)0, acc, false, false);
        }
        const float bias = b2[n0 + l16];
        #pragma unroll
        for (int v = 0; v < 8; ++v) {
            const int m = m0 + v + 8 * half;
            g[(size_t)m * NC + n0 + l16] = sigmoidf_(acc[v] + bias);
        }
    }
}

// ---------------------------------------------------------------------------
// Kernel 3: gate. Reverse traversal order -> the tail of x is still resident
// in the 192MB L2 after the pool pass; read it first. Non-temporal stores so
// the streamed output does not evict x from L2.
// ---------------------------------------------------------------------------
__global__ __launch_bounds__(256) void se_gate(const float* __restrict__ x,
                                               const float* __restrict__ g,
                                               float* __restrict__ out, int n4) {
    const int i = blockIdx.x * 256 + threadIdx.x;
    if (i >= n4) return;
    const int j = n4 - 1 - i;          // reverse order (still 128B-contiguous per block)
    const int plane = j / HW4;
    const float gv = g[plane];
    v4f a = ((const v4f*)x)[j];
    a.x *= gv; a.y *= gv; a.z *= gv; a.w *= gv;
    __builtin_nontemporal_store(a, ((v4f*)out) + j);
}

// ---------------------------------------------------------------------------
extern "C" void kernel_launch(void* const* d_in, const int* in_sizes, int n_in,
                              void* d_out, int out_size, void* d_ws, size_t ws_size,
                              hipStream_t stream) {
    const float* x  = (const float*)d_in[0];   // [64,384,56,56]
    const float* w1 = (const float*)d_in[1];   // [96,384]
    const float* b1 = (const float*)d_in[2];   // [96]
    const float* w2 = (const float*)d_in[3];   // [384,96]
    const float* b2 = (const float*)d_in[4];   // [384]
    float* out = (float*)d_out;

    float* y = (float*)d_ws;         // [64,384] pooled means
    float* g = y + NPLANES;          // [64,384] gates

    se_pool<<<NPLANES, 256, 0, stream>>>(x, y);
    se_mlp<<<1, 1024, 0, stream>>>(y, w1, b1, w2, b2, g);

    const int n4 = NPLANES * HW4;    // 19,267,584 float4
    se_gate<<<(n4 + 255) / 256, 256, 0, stream>>>(x, g, out, n4);
}